// finalAttention_35742717837751
// MI455X (gfx1250) — compile-verified
//
#include <hip/hip_runtime.h>
#include <cstdint>

typedef float f4  __attribute__((ext_vector_type(4)));
typedef float v2f __attribute__((ext_vector_type(2)));
typedef float v8f __attribute__((ext_vector_type(8)));

#define T_DIM   4096
#define B_DIM   32
#define H_DIM   512
#define C_DIM   32000
#define TCHUNK  64
#define NCHUNK  (T_DIM / TCHUNK)   // 64 chunks -> 2048 waves in pass 1

// ---------------------------------------------------------------------------
// Pass 1: fused alignment-score + softmax-weighted accumulation (one read of x)
// wave w handles batch b = w/NCHUNK, timesteps [chunk*TCHUNK, +TCHUNK)
// lane holds h = {i*128 + lane*4 + j : i<4, j<4} (coalesced float4s)
// ---------------------------------------------------------------------------
__global__ __launch_bounds__(256) void attn_pass1(
    const float* __restrict__ x, const float* __restrict__ aw,
    const float* __restrict__ ab, float* __restrict__ pacc,
    float* __restrict__ ps)
{
  const int wid  = (blockIdx.x * blockDim.x + threadIdx.x) >> 5;
  const int lane = threadIdx.x & 31;
  const int b     = wid / NCHUNK;
  const int chunk = wid % NCHUNK;
  const int t0    = chunk * TCHUNK;
  const float alignb = ab[0];

  f4 w[4];
  #pragma unroll
  for (int i = 0; i < 4; ++i)
    w[i] = *(const f4*)(aw + i * 128 + lane * 4);

  f4 acc[4] = {};
  float s = 0.0f;

  for (int t = 0; t < TCHUNK; ++t) {
    const float* p = x + ((size_t)(t0 + t) * B_DIM + b) * H_DIM;
    f4 v[4];
    #pragma unroll
    for (int i = 0; i < 4; ++i)   // streaming 256MB: non-temporal, keep L2 for out_w
      v[i] = __builtin_nontemporal_load((const f4*)(p + i * 128 + lane * 4));

    float d = 0.0f;
    #pragma unroll
    for (int i = 0; i < 4; ++i)
      d += v[i].x * w[i].x + v[i].y * w[i].y + v[i].z * w[i].z + v[i].w * w[i].w;

    #pragma unroll
    for (int m = 16; m > 0; m >>= 1)      // wave32 reduction
      d += __shfl_xor(d, m, 32);

    float e    = tanhf(d + alignb);
    float pexp = __expf(e - 1.0f);        // tanh<=1: shift makes exp safe, softmax invariant
    s += pexp;
    #pragma unroll
    for (int i = 0; i < 4; ++i)
      acc[i] += pexp * v[i];
  }

  float* dst = pacc + (size_t)wid * H_DIM;
  #pragma unroll
  for (int i = 0; i < 4; ++i)
    *(f4*)(dst + i * 128 + lane * 4) = acc[i];
  if (lane == 0) ps[wid] = s;
}

// ---------------------------------------------------------------------------
// Pass 2: combine NCHUNK partials per batch, normalize -> context [B,H]
// ---------------------------------------------------------------------------
__global__ __launch_bounds__(128) void attn_pass2(
    const float* __restrict__ pacc, const float* __restrict__ ps,
    float* __restrict__ ctx)
{
  const int b   = blockIdx.x;
  const int tid = threadIdx.x;            // 128 threads * 4 h = 512
  float s = 0.0f;
  for (int c = 0; c < NCHUNK; ++c) s += ps[b * NCHUNK + c];
  f4 a = {};
  for (int c = 0; c < NCHUNK; ++c)
    a += *(const f4*)(pacc + ((size_t)(b * NCHUNK + c)) * H_DIM + tid * 4);
  const float inv = 1.0f / s;
  *(f4*)(ctx + b * H_DIM + tid * 4) = a * inv;
}

// ---------------------------------------------------------------------------
// Pass 3: out[b,c] = context[b,:] . out_w[c,:] + out_b[c] via V_WMMA_F32_16X16X4_F32
// Each wave: one 16(M) x 64(N) strip, 4 accumulators, A reused, K-loop of 128.
// A layout (16x4 f32): lane<16 -> (M=lane, K=0..1); lane>=16 -> (M=lane-16, K=2..3)
// B layout (4x16 f32): mirrored; C/D: v8f, M = v + 8*(lane>=16), N = lane&15
// ---------------------------------------------------------------------------
__global__ __launch_bounds__(256) void attn_gemm(
    const float* __restrict__ ctx, const float* __restrict__ w,
    const float* __restrict__ bias, float* __restrict__ out)
{
  const int lane = threadIdx.x & 31;
  const int gw   = (blockIdx.x * blockDim.x + threadIdx.x) >> 5;  // 0..999
  const int m0   = (gw & 1) * 16;
  const int n0   = (gw >> 1) * 64;

  const int mrow = m0 + (lane & 15);
  const int ncol = lane & 15;
  const int kb   = (lane >> 4) * 2;      // upper half-wave holds K+2,K+3

  const float* arow  = ctx + (size_t)mrow * H_DIM + kb;
  const float* brow0 = w + (size_t)(n0 + ncol) * H_DIM + kb;

  v8f acc0 = {}, acc1 = {}, acc2 = {}, acc3 = {};

  for (int k = 0; k < H_DIM; k += 4) {
    v2f a = *(const v2f*)(arow + k);
    v2f b0 = *(const v2f*)(brow0 + (size_t)0 * 16 * H_DIM + k);
    v2f b1 = *(const v2f*)(brow0 + (size_t)1 * 16 * H_DIM + k);
    v2f b2 = *(const v2f*)(brow0 + (size_t)2 * 16 * H_DIM + k);
    v2f b3 = *(const v2f*)(brow0 + (size_t)3 * 16 * H_DIM + k);
    acc0 = __builtin_amdgcn_wmma_f32_16x16x4_f32(false, a, false, b0, (short)0, acc0, false, false);
    acc1 = __builtin_amdgcn_wmma_f32_16x16x4_f32(false, a, false, b1, (short)0, acc1, false, false);
    acc2 = __builtin_amdgcn_wmma_f32_16x16x4_f32(false, a, false, b2, (short)0, acc2, false, false);
    acc3 = __builtin_amdgcn_wmma_f32_16x16x4_f32(false, a, false, b3, (short)0, acc3, false, false);
  }

  const int half = lane >> 4;
  v8f accs[4] = {acc0, acc1, acc2, acc3};
  #pragma unroll
  for (int j = 0; j < 4; ++j) {
    const int c  = n0 + j * 16 + ncol;
    const float bb = bias[c];
    #pragma unroll
    for (int v = 0; v < 8; ++v) {
      const int row = m0 + v + 8 * half;
      out[(size_t)row * C_DIM + c] = accs[j][v] + bb;
    }
  }
}

// ---------------------------------------------------------------------------
extern "C" void kernel_launch(void* const* d_in, const int* in_sizes, int n_in,
                              void* d_out, int out_size, void* d_ws, size_t ws_size,
                              hipStream_t stream) {
  const float* x  = (const float*)d_in[0];   // [T,B,H]
  const float* aw = (const float*)d_in[1];   // [H]
  const float* ab = (const float*)d_in[2];   // scalar
  const float* ow = (const float*)d_in[3];   // [C,H]
  const float* ob = (const float*)d_in[4];   // [C]
  float* out = (float*)d_out;                // [B,C]

  float* pacc = (float*)d_ws;                              // B*NCHUNK*H   (4 MB)
  float* ps   = pacc + (size_t)B_DIM * NCHUNK * H_DIM;     // B*NCHUNK    (8 KB)
  float* ctx  = ps + B_DIM * NCHUNK;                       // B*H         (64 KB)

  // 2048 waves = 65536 threads
  attn_pass1<<<(B_DIM * NCHUNK * 32) / 256, 256, 0, stream>>>(x, aw, ab, pacc, ps);
  attn_pass2<<<B_DIM, 128, 0, stream>>>(pacc, ps, ctx);
  // 2 M-tiles * 500 N-strips = 1000 waves = 125 blocks (all waves full -> EXEC all 1s)
  attn_gemm<<<(2 * (C_DIM / 64)) / 8, 256, 0, stream>>>(ctx, ow, ob, out);
}